// Model_67774583931143
// MI455X (gfx1250) — compile-verified
//
#include <hip/hip_runtime.h>
#include <hip/hip_bf16.h>
#include <float.h>
#include <math.h>

typedef __attribute__((ext_vector_type(2))) float v2f;
typedef __attribute__((ext_vector_type(8))) float v8f;

#define N_NODES  200000
#define N_EDGES  3200000
#define N_GRAPHS 2000
#define NMAX     256
#define KTOP     30
#define HDIM     97      // 32+32+32+1 concatenated features

// ---------------------------------------------------------------------------
// WMMA f32 GEMM: C[M,N] = A[M,K] @ B[K,N] (+bias, +relu). One wave per 16x16
// output tile, K stepped by 4 through V_WMMA_F32_16X16X4_F32.
// A-frag (16x4): lane L holds row M=L%16; vgpr v holds K = 2*(L/16)+v.
// B-frag (4x16): lane L holds col N=L%16; vgpr v holds K = 2*(L/16)+v.
// C/D (16x16):   vgpr r -> M = r + 8*(L/16), N = L%16.
// ---------------------------------------------------------------------------
__global__ void gemm_wmma_f32(const float* __restrict__ A, int lda,
                              const float* __restrict__ B, int ldb,
                              float* __restrict__ C, int ldc,
                              int M, int Kdim,
                              const float* __restrict__ bias, int relu)
{
    const int lane  = threadIdx.x;                       // 0..31
    const int mtile = blockIdx.x * blockDim.y + threadIdx.y;
    const int ntile = blockIdx.y;
    if (mtile * 16 >= M) return;                         // uniform per wave

    const int r  = lane & 15;
    const int g2 = (lane >> 4) << 1;                     // 0 or 2
    const int m0 = mtile * 16;
    const int n0 = ntile * 16;

    const float* Arow = A + (size_t)(m0 + r) * lda;
    const float* Bcol = B + (n0 + r);

    v8f acc = {};
    for (int k0 = 0; k0 < Kdim; k0 += 4) {
        v2f a, b;
        a.x = Arow[k0 + g2];
        a.y = Arow[k0 + g2 + 1];
        b.x = Bcol[(size_t)(k0 + g2) * ldb];
        b.y = Bcol[(size_t)(k0 + g2 + 1) * ldb];
        acc = __builtin_amdgcn_wmma_f32_16x16x4_f32(
                  /*neg_a=*/false, a, /*neg_b=*/false, b,
                  /*c_mod=*/(short)0, acc, /*reuse_a=*/false, /*reuse_b=*/false);
    }

    const int row_off = (lane >> 4) << 3;                // +8 for lanes 16..31
    #pragma unroll
    for (int i = 0; i < 8; ++i) {
        int row = m0 + row_off + i;
        int col = n0 + r;
        float v = acc[i];
        if (bias) v += bias[col];
        if (relu) v = fmaxf(v, 0.f);
        C[(size_t)row * ldc + col] = v;
    }
}

// ---------------------------------------------------------------------------
// GCN helpers
// ---------------------------------------------------------------------------
__global__ void zero_f32(float* __restrict__ p, size_t n)
{
    size_t i = (size_t)blockIdx.x * blockDim.x + threadIdx.x;
    if (i < n) p[i] = 0.f;
}

__global__ void deg_edges(const int* __restrict__ src, const int* __restrict__ dst,
                          float* __restrict__ deg)
{
    int e = blockIdx.x * blockDim.x + threadIdx.x;
    if (e >= N_EDGES) return;
    int s = src[e], d = dst[e];
    if (s != d) atomicAdd(&deg[d], 1.0f);
}

__global__ void make_dis(float* __restrict__ deg)   // deg -> rsqrt(deg+1)
{
    int v = blockIdx.x * blockDim.x + threadIdx.x;
    if (v < N_NODES) deg[v] = rsqrtf(deg[v] + 1.0f);
}

// agg = hw * dis^2 (self term), Dout=32 vectorized: one thread = one float4
__global__ void self_init32(const float4* __restrict__ hw, const float* __restrict__ dis,
                            float4* __restrict__ agg)
{
    size_t i = (size_t)blockIdx.x * blockDim.x + threadIdx.x;   // over N*8 float4s
    if (i >= (size_t)N_NODES * 8) return;
    int v = (int)(i >> 3);
    float d = dis[v];
    float d2 = d * d;
    float4 h = hw[i];
    float4 o = make_float4(h.x * d2, h.y * d2, h.z * d2, h.w * d2);
    agg[i] = o;
}

__global__ void self_init1(const float* __restrict__ hw, const float* __restrict__ dis,
                           float* __restrict__ agg)
{
    int v = blockIdx.x * blockDim.x + threadIdx.x;
    if (v >= N_NODES) return;
    float d = dis[v];
    agg[v] = hw[v] * d * d;
}

// agg[dst] += hw[src] * dis[src]*dis[dst]  (src != dst), Dout=32.
// float4 gathers (global_load_b128) + f32 atomics.
__global__ void edge_agg32(const int* __restrict__ src, const int* __restrict__ dst,
                           const float* __restrict__ dis, const float4* __restrict__ hw,
                           float* __restrict__ agg)
{
    int e = blockIdx.x * blockDim.x + threadIdx.x;
    if (e >= N_EDGES) return;
    int s = src[e], d = dst[e];
    if (s == d) return;
    float w = dis[s] * dis[d];
    const float4* hs = hw + (size_t)s * 8;      // 32 floats = 8 float4
    float* ad = agg + (size_t)d * 32;
    #pragma unroll
    for (int q = 0; q < 8; ++q) {
        float4 h = hs[q];
        atomicAdd(&ad[4 * q + 0], h.x * w);
        atomicAdd(&ad[4 * q + 1], h.y * w);
        atomicAdd(&ad[4 * q + 2], h.z * w);
        atomicAdd(&ad[4 * q + 3], h.w * w);
    }
}

__global__ void edge_agg1(const int* __restrict__ src, const int* __restrict__ dst,
                          const float* __restrict__ dis, const float* __restrict__ hw,
                          float* __restrict__ agg)
{
    int e = blockIdx.x * blockDim.x + threadIdx.x;
    if (e >= N_EDGES) return;
    int s = src[e], d = dst[e];
    if (s == d) return;
    atomicAdd(&agg[d], hw[s] * dis[s] * dis[d]);
}

// Hcat[v, colofs + c] = tanh(agg[v,c] + bias[c])
__global__ void finalize_tanh(const float* __restrict__ agg, const float* __restrict__ bias,
                              float* __restrict__ Hcat, int colofs, int Dout)
{
    size_t i = (size_t)blockIdx.x * blockDim.x + threadIdx.x;
    if (i >= (size_t)N_NODES * Dout) return;
    int v = (int)(i / Dout);
    int c = (int)(i % Dout);
    Hcat[(size_t)v * HDIM + colofs + c] = tanhf(agg[i] + bias[c]);
}

// hw[v] = dot(Hcat[v, colofs..colofs+31], W[0..31])   (layer-4 GEMV, Dout=1)
__global__ void gemv_32to1(const float* __restrict__ Hcat, int colofs,
                           const float* __restrict__ W, float* __restrict__ out)
{
    int v = blockIdx.x * blockDim.x + threadIdx.x;
    if (v >= N_NODES) return;
    const float* h = Hcat + (size_t)v * HDIM + colofs;
    float s = 0.f;
    #pragma unroll
    for (int c = 0; c < 32; ++c) s += h[c] * W[c];
    out[v] = s;
}

// ---------------------------------------------------------------------------
// SortPool
// ---------------------------------------------------------------------------
__global__ void graph_offsets(const int* __restrict__ batch,
                              int* __restrict__ off, int* __restrict__ cnt)
{
    int g = blockIdx.x * blockDim.x + threadIdx.x;
    if (g >= N_GRAPHS) return;
    int lo, hi;
    lo = 0; hi = N_NODES;
    while (lo < hi) { int mid = (lo + hi) >> 1; if (batch[mid] < g) lo = mid + 1; else hi = mid; }
    int a = lo;
    lo = 0; hi = N_NODES;
    while (lo < hi) { int mid = (lo + hi) >> 1; if (batch[mid] < g + 1) lo = mid + 1; else hi = mid; }
    off[g] = a;
    cnt[g] = lo - a;
}

// One block (256 threads) per graph: bitonic-sort keys descending, gather top-K.
__global__ void sortpool(const float* __restrict__ Hcat,
                         const int* __restrict__ off, const int* __restrict__ cnt,
                         float* __restrict__ pooled)
{
    __shared__ float skey[NMAX];
    __shared__ int   sidx[NMAX];
    const int g = blockIdx.x;
    const int tid = threadIdx.x;
    const int o = off[g];
    int n = cnt[g]; if (n > NMAX) n = NMAX;    // positions >= 256 are dropped

    skey[tid] = (tid < n) ? Hcat[(size_t)(o + tid) * HDIM + 96] : -FLT_MAX;
    sidx[tid] = tid;
    __syncthreads();

    for (int k = 2; k <= NMAX; k <<= 1) {
        for (int j = k >> 1; j > 0; j >>= 1) {
            int i = tid, ixj = tid ^ j;
            if (ixj > i) {
                bool desc = ((i & k) == 0);
                float ki = skey[i], kj = skey[ixj];
                bool sw = desc ? (ki < kj) : (ki > kj);
                if (sw) {
                    skey[i] = kj; skey[ixj] = ki;
                    int t = sidx[i]; sidx[i] = sidx[ixj]; sidx[ixj] = t;
                }
            }
            __syncthreads();
        }
    }

    for (int z = tid; z < KTOP * HDIM; z += blockDim.x) {
        int t = z / HDIM, j = z % HDIM;
        float v = 0.f;                               // padded slots -> 0 (fill rule)
        if (t < n) v = Hcat[(size_t)(o + sidx[t]) * HDIM + j];
        pooled[(size_t)g * (KTOP * HDIM) + z] = v;
    }
}

// ---------------------------------------------------------------------------
// Head: conv(97,stride97)+relu+pairmax, conv(5)+relu, fc2 + log_softmax
// ---------------------------------------------------------------------------
__global__ void conv5_pool(const float* __restrict__ pooled,
                           const float* __restrict__ cw5, const float* __restrict__ cb5,
                           float* __restrict__ m5)
{
    int idx = blockIdx.x * blockDim.x + threadIdx.x;   // g*16*15 + c*15 + u
    if (idx >= N_GRAPHS * 16 * 15) return;
    int u = idx % 15;
    int c = (idx / 15) % 16;
    int g = idx / (15 * 16);
    const float* w = cw5 + c * HDIM;
    float m = -FLT_MAX;
    for (int p = 0; p < 2; ++p) {
        int t = 2 * u + p;
        const float* x = pooled + (size_t)g * (KTOP * HDIM) + t * HDIM;
        float s = cb5[c];
        for (int j = 0; j < HDIM; ++j) s += w[j] * x[j];
        m = fmaxf(m, s);
    }
    m5[idx] = fmaxf(m, 0.f);                           // max(relu(a),relu(b)) == relu(max)
}

__global__ void conv6(const float* __restrict__ m5,
                      const float* __restrict__ cw6, const float* __restrict__ cb6,
                      float* __restrict__ y6)
{
    int idx = blockIdx.x * blockDim.x + threadIdx.x;   // over g*32*11
    if (idx >= N_GRAPHS * 32 * 11) return;
    int t = idx % 11;
    int o = (idx / 11) % 32;
    int g = idx / (11 * 32);
    float s = cb6[o];
    for (int i = 0; i < 16; ++i) {
        const float* mm = m5 + (size_t)g * 240 + i * 15 + t;
        const float* w  = cw6 + o * 80 + i * 5;
        #pragma unroll
        for (int k = 0; k < 5; ++k) s += w[k] * mm[k];
    }
    y6[(size_t)g * 352 + o * 11 + t] = fmaxf(s, 0.f);
}

__global__ void fc2_kernel(const float* __restrict__ r1,
                           const float* __restrict__ fw2, const float* __restrict__ fb2,
                           float* __restrict__ logit)
{
    int idx = blockIdx.x * blockDim.x + threadIdx.x;   // g*10 + c
    if (idx >= N_GRAPHS * 10) return;
    int c = idx % 10, g = idx / 10;
    const float* rr = r1 + (size_t)g * 128;
    float s = fb2[c];
    for (int j = 0; j < 128; ++j) s += rr[j] * fw2[j * 10 + c];
    logit[idx] = s;
}

__global__ void logsoftmax(const float* __restrict__ logit, float* __restrict__ out)
{
    int g = blockIdx.x * blockDim.x + threadIdx.x;
    if (g >= N_GRAPHS) return;
    const float* l = logit + g * 10;
    float m = -FLT_MAX;
    for (int c = 0; c < 10; ++c) m = fmaxf(m, l[c]);
    float s = 0.f;
    for (int c = 0; c < 10; ++c) s += expf(l[c] - m);
    float lse = m + logf(s);
    for (int c = 0; c < 10; ++c) out[g * 10 + c] = l[c] - lse;
}

// ---------------------------------------------------------------------------
// Launcher
// ---------------------------------------------------------------------------
extern "C" void kernel_launch(void* const* d_in, const int* in_sizes, int n_in,
                              void* d_out, int out_size, void* d_ws, size_t ws_size,
                              hipStream_t stream)
{
    const float* x     = (const float*)d_in[0];
    const int*   ei    = (const int*)  d_in[1];
    const int*   batch = (const int*)  d_in[2];
    const float* W1 = (const float*)d_in[3];  const float* b1 = (const float*)d_in[4];
    const float* W2 = (const float*)d_in[5];  const float* b2 = (const float*)d_in[6];
    const float* W3 = (const float*)d_in[7];  const float* b3 = (const float*)d_in[8];
    const float* W4 = (const float*)d_in[9];  const float* b4 = (const float*)d_in[10];
    const float* cw5 = (const float*)d_in[11]; const float* cb5 = (const float*)d_in[12];
    const float* cw6 = (const float*)d_in[13]; const float* cb6 = (const float*)d_in[14];
    const float* fw1 = (const float*)d_in[15]; const float* fb1 = (const float*)d_in[16];
    const float* fw2 = (const float*)d_in[17]; const float* fb2 = (const float*)d_in[18];
    float* out = (float*)d_out;

    const int* src = ei;
    const int* dst = ei + N_EDGES;

    // ---- workspace carve (float4-aligned: d_ws is allocation-aligned) ----
    float* p = (float*)d_ws;
    float* dis    = p;  p += N_NODES;                    // deg -> rsqrt(deg+1)
    p += 8;                                              // keep hw 16B-aligned
    float* hw     = p;  p += (size_t)N_NODES * 32;
    float* agg    = p;  p += (size_t)N_NODES * 32;
    float* Hcat   = p;  p += (size_t)N_NODES * HDIM;
    float* pooled = p;  p += (size_t)N_GRAPHS * KTOP * HDIM;
    float* m5     = p;  p += (size_t)N_GRAPHS * 240;
    float* y6     = p;  p += (size_t)N_GRAPHS * 352;
    float* r1     = p;  p += (size_t)N_GRAPHS * 128;
    float* logit  = p;  p += (size_t)N_GRAPHS * 10;
    int*   off    = (int*)p;
    int*   cnt    = off + N_GRAPHS;

    const int TB = 256;
    const int edgeBlocks = (N_EDGES + TB - 1) / TB;
    const int nodeBlocks = (N_NODES + TB - 1) / TB;
    const size_t nc32 = (size_t)N_NODES * 32;
    const int nc32Blocks = (int)((nc32 + TB - 1) / TB);
    const int nv4Blocks  = (int)(((size_t)N_NODES * 8 + TB - 1) / TB);

    // ---- degrees / normalization ----
    zero_f32<<<nodeBlocks, TB, 0, stream>>>(dis, N_NODES);
    deg_edges<<<edgeBlocks, TB, 0, stream>>>(src, dst, dis);
    make_dis<<<nodeBlocks, TB, 0, stream>>>(dis);

    dim3 gemmBlock(32, 4);   // 4 waves, one 16x16 tile each

    // ---- GCN layer 1: x(200000x128) @ W1(128x32) ----
    {
        dim3 grid((N_NODES / 16 + 3) / 4, 32 / 16);
        gemm_wmma_f32<<<grid, gemmBlock, 0, stream>>>(x, 128, W1, 32, hw, 32,
                                                      N_NODES, 128, nullptr, 0);
        self_init32<<<nv4Blocks, TB, 0, stream>>>((const float4*)hw, dis, (float4*)agg);
        edge_agg32<<<edgeBlocks, TB, 0, stream>>>(src, dst, dis, (const float4*)hw, agg);
        finalize_tanh<<<nc32Blocks, TB, 0, stream>>>(agg, b1, Hcat, 0, 32);
    }
    // ---- GCN layers 2,3: Hcat slice (stride 97) @ W(32x32) ----
    {
        dim3 grid((N_NODES / 16 + 3) / 4, 32 / 16);
        gemm_wmma_f32<<<grid, gemmBlock, 0, stream>>>(Hcat + 0, HDIM, W2, 32, hw, 32,
                                                      N_NODES, 32, nullptr, 0);
        self_init32<<<nv4Blocks, TB, 0, stream>>>((const float4*)hw, dis, (float4*)agg);
        edge_agg32<<<edgeBlocks, TB, 0, stream>>>(src, dst, dis, (const float4*)hw, agg);
        finalize_tanh<<<nc32Blocks, TB, 0, stream>>>(agg, b2, Hcat, 32, 32);

        gemm_wmma_f32<<<grid, gemmBlock, 0, stream>>>(Hcat + 32, HDIM, W3, 32, hw, 32,
                                                      N_NODES, 32, nullptr, 0);
        self_init32<<<nv4Blocks, TB, 0, stream>>>((const float4*)hw, dis, (float4*)agg);
        edge_agg32<<<edgeBlocks, TB, 0, stream>>>(src, dst, dis, (const float4*)hw, agg);
        finalize_tanh<<<nc32Blocks, TB, 0, stream>>>(agg, b3, Hcat, 64, 32);
    }
    // ---- GCN layer 4 (Dout = 1) ----
    {
        gemv_32to1<<<nodeBlocks, TB, 0, stream>>>(Hcat, 64, W4, hw);
        self_init1<<<nodeBlocks, TB, 0, stream>>>(hw, dis, agg);
        edge_agg1<<<edgeBlocks, TB, 0, stream>>>(src, dst, dis, hw, agg);
        finalize_tanh<<<nodeBlocks, TB, 0, stream>>>(agg, b4, Hcat, 96, 1);
    }

    // ---- SortPool ----
    graph_offsets<<<(N_GRAPHS + TB - 1) / TB, TB, 0, stream>>>(batch, off, cnt);
    sortpool<<<N_GRAPHS, NMAX, 0, stream>>>(Hcat, off, cnt, pooled);

    // ---- 1-D conv head ----
    conv5_pool<<<(N_GRAPHS * 16 * 15 + TB - 1) / TB, TB, 0, stream>>>(pooled, cw5, cb5, m5);
    conv6<<<(N_GRAPHS * 32 * 11 + TB - 1) / TB, TB, 0, stream>>>(m5, cw6, cb6, y6);

    // ---- fc1 via WMMA (2000x352 @ 352x128, fused bias+relu) ----
    {
        dim3 grid((N_GRAPHS / 16 + 3) / 4, 128 / 16);
        gemm_wmma_f32<<<grid, gemmBlock, 0, stream>>>(y6, 352, fw1, 128, r1, 128,
                                                      N_GRAPHS, 352, fb1, 1);
    }

    // ---- fc2 + log_softmax ----
    fc2_kernel<<<(N_GRAPHS * 10 + TB - 1) / TB, TB, 0, stream>>>(r1, fw2, fb2, logit);
    logsoftmax<<<(N_GRAPHS + TB - 1) / TB, TB, 0, stream>>>(logit, out);
}